// ComposeTransform_44341242364449
// MI455X (gfx1250) — compile-verified
//
#include <hip/hip_runtime.h>
#include <hip/hip_bf16.h>

// ComposeTransform: out = trilinear(d1, grid + d2) + d2, batched.
// Shapes: [2, 160, 192, 160, 3] float32. Pure HBM-bound gather kernel.

#define DD 160
#define HH 192
#define WW 160
#define NVOX (DD * HH * WW)            // 4,915,200 voxels per batch
#define SLICE (HH * WW)                // 30,720 voxels per x-slice (= 120 * 256)

__global__ __launch_bounds__(256) void compose_disp_kernel(
    const float* __restrict__ d1,      // [2, D, H, W, 3]
    const float* __restrict__ d2,      // [2, D, H, W, 3]
    float* __restrict__ out)           // [2, D, H, W, 3]
{
  // Exact tiling: gridDim = (SLICE/256, DD, 2); no bounds check needed.
  const int p  = blockIdx.x * 256 + threadIdx.x;   // 0 .. SLICE-1
  const int iy = p / WW;                           // const div -> mulhi
  const int iz = p - iy * WW;
  const int ix = blockIdx.y;
  const int b  = blockIdx.z;

  const int r = (ix * HH + iy) * WW + iz;          // voxel index in batch
  const float* __restrict__ d1b = d1 + (size_t)b * (NVOX * 3);
  const float* __restrict__ d2p = d2 + (size_t)b * (NVOX * 3) + (size_t)r * 3;
  float*       __restrict__ op  = out + (size_t)b * (NVOX * 3) + (size_t)r * 3;

  // Streaming read of d2: non-temporal (no reuse; keep L2 for d1 gather lines).
  const float ux = __builtin_nontemporal_load(d2p + 0);
  const float uy = __builtin_nontemporal_load(d2p + 1);
  const float uz = __builtin_nontemporal_load(d2p + 2);

  // Absolute sample location = grid + displacement
  const float px = (float)ix + ux;
  const float py = (float)iy + uy;
  const float pz = (float)iz + uz;

  // Reference semantics: clip floor and ceil corners to volume, weight for the
  // floor corner = clip(clipped_ceil - loc, 0, 1).
  const float fx = floorf(px), fy = floorf(py), fz = floorf(pz);

  const float x0f = fminf(fmaxf(fx,        0.f), (float)(DD - 1));
  const float x1f = fminf(fmaxf(fx + 1.f,  0.f), (float)(DD - 1));
  const float y0f = fminf(fmaxf(fy,        0.f), (float)(HH - 1));
  const float y1f = fminf(fmaxf(fy + 1.f,  0.f), (float)(HH - 1));
  const float z0f = fminf(fmaxf(fz,        0.f), (float)(WW - 1));
  const float z1f = fminf(fmaxf(fz + 1.f,  0.f), (float)(WW - 1));

  const float wxf = fminf(fmaxf(x1f - px, 0.f), 1.f);  // floor-corner weight
  const float wyf = fminf(fmaxf(y1f - py, 0.f), 1.f);
  const float wzf = fminf(fmaxf(z1f - pz, 0.f), 1.f);
  const float wxc = 1.f - wxf;
  const float wyc = 1.f - wyf;
  const float wzc = 1.f - wzf;

  const int x0 = (int)x0f, x1 = (int)x1f;
  const int y0 = (int)y0f, y1 = (int)y1f;
  const int z0 = (int)z0f, z1 = (int)z1f;

  // 8 corner flat offsets (in floats), layout [D,H,W,3]
  const int xr0 = x0 * HH, xr1 = x1 * HH;
  const int a00 = (xr0 + y0) * WW;
  const int a01 = (xr0 + y1) * WW;
  const int a10 = (xr1 + y0) * WW;
  const int a11 = (xr1 + y1) * WW;

  const int f000 = (a00 + z0) * 3;
  const int f001 = (a00 + z1) * 3;
  const int f010 = (a01 + z0) * 3;
  const int f011 = (a01 + z1) * 3;
  const int f100 = (a10 + z0) * 3;
  const int f101 = (a10 + z1) * 3;
  const int f110 = (a11 + z0) * 3;
  const int f111 = (a11 + z1) * 3;

  // Trilinear weights (x-major corner naming wXYZ, f=floor c=ceil)
  const float wff = wxf * wyf, wfc = wxf * wyc;
  const float wcf = wxc * wyf, wcc = wxc * wyc;
  const float w000 = wff * wzf, w001 = wff * wzc;
  const float w010 = wfc * wzf, w011 = wfc * wzc;
  const float w100 = wcf * wzf, w101 = wcf * wzc;
  const float w110 = wcf * 0.f + wcc * wzf, w111 = wcc * wzc;

  float r0 = 0.f, r1 = 0.f, r2 = 0.f;
#define ACCUM(F, W)                                   \
  {                                                   \
    const float* __restrict__ cp = d1b + (F);         \
    r0 = fmaf((W), cp[0], r0);                        \
    r1 = fmaf((W), cp[1], r1);                        \
    r2 = fmaf((W), cp[2], r2);                        \
  }
  ACCUM(f000, w000)
  ACCUM(f001, w001)
  ACCUM(f010, w010)
  ACCUM(f011, w011)
  ACCUM(f100, w100)
  ACCUM(f101, w101)
  ACCUM(f110, w110)
  ACCUM(f111, w111)
#undef ACCUM

  // Streaming write: non-temporal store (gfx1250 TH_NT policy).
  __builtin_nontemporal_store(r0 + ux, op + 0);
  __builtin_nontemporal_store(r1 + uy, op + 1);
  __builtin_nontemporal_store(r2 + uz, op + 2);
}

extern "C" void kernel_launch(void* const* d_in, const int* in_sizes, int n_in,
                              void* d_out, int out_size, void* d_ws, size_t ws_size,
                              hipStream_t stream) {
  const float* d1 = (const float*)d_in[0];  // disp_1
  const float* d2 = (const float*)d_in[1];  // disp_2
  float* out = (float*)d_out;

  dim3 grid(SLICE / 256, DD, 2);   // (120, 160, 2)
  dim3 block(256, 1, 1);           // 8 wave32s per block
  compose_disp_kernel<<<grid, block, 0, stream>>>(d1, d2, out);
}